// CFConv_81827716923574
// MI455X (gfx1250) — compile-verified
//
#include <hip/hip_runtime.h>

typedef __attribute__((ext_vector_type(2))) float v2f;
typedef __attribute__((ext_vector_type(4))) float v4f;
typedef __attribute__((ext_vector_type(8))) float v8f;

#define FAN 128

// ---------------------------------------------------------------------------
// Zero-fill (conv accumulator must start at 0; harness poisons ws with 0xAA)
// ---------------------------------------------------------------------------
__global__ __launch_bounds__(256) void zero_kernel(float* __restrict__ p, long long n) {
  long long i = (long long)blockIdx.x * blockDim.x + threadIdx.x;
  const long long stride = (long long)gridDim.x * blockDim.x;
  for (; i < n; i += stride) p[i] = 0.0f;
}

// ---------------------------------------------------------------------------
// Y[num_rows x 128] = A[num_rows x 128] * W[128 x 128] (+ bias), fp32 WMMA.
// One wave per 16-wide N-tile (8 waves = 256 threads cover all 128 cols).
// B fragments for the wave's tile (32 K-steps x v2f = 64 VGPRs) are loaded
// once and reused across a grid-stride loop over 16-row blocks. The K loop is
// FULLY unrolled so Bf[] is register-promoted (partial unroll previously
// demoted it to scratch -> scratch_load_b128 per WMMA).
//
// V_WMMA_F32_16X16X4_F32 operand layout (ISA 7.12.2):
//   A 16x4 : lanes 0-15 -> M=lane, {v0,v1} = K0,K1 ; lanes 16-31 -> K2,K3
//   B 4x16 : lanes 0-15 -> N=lane, {v0,v1} = K0,K1 ; lanes 16-31 -> K2,K3
//   C/D    : VGPR r -> row r (lanes 0-15) / row r+8 (lanes 16-31), N = lane&15
// ---------------------------------------------------------------------------
__global__ __launch_bounds__(256) void gemm128_wmma_kernel(
    const float* __restrict__ A,
    const float* __restrict__ W,
    const float* __restrict__ bias,   // may be null
    float* __restrict__ Y,
    int num_rows)
{
  const int lane = threadIdx.x & 31;
  const int nt   = threadIdx.x >> 5;      // wave id == N-tile (0..7)
  const int m    = lane & 15;
  const int kh   = (lane >> 4) << 1;      // 0 or 2 (K sub-offset per half-wave)
  const int ncol = (nt << 4) + m;

  // Preload this tile's B fragments (whole K=128): 32 x v2f, kept in VGPRs.
  v2f Bf[32];
#pragma unroll
  for (int ks = 0; ks < 32; ++ks) {
    const float* bp = W + (size_t)(ks * 4 + kh) * FAN + ncol;
    v2f b; b.x = bp[0]; b.y = bp[FAN];
    Bf[ks] = b;
  }

  const float cinit = bias ? bias[ncol] : 0.0f;
  const int rowBlocks = (num_rows + 15) >> 4;
  const int mh = (lane >> 4) << 3;        // 0 or 8

  for (int blk = blockIdx.x; blk < rowBlocks; blk += gridDim.x) {
    const int rowBase = blk << 4;
    int mrow = rowBase + m;
    if (mrow >= num_rows) mrow = num_rows - 1;       // tail clamp (loads)
    const float* ap = A + (size_t)mrow * FAN + kh;

    v8f c = {cinit, cinit, cinit, cinit, cinit, cinit, cinit, cinit};
#pragma unroll
    for (int ks = 0; ks < 32; ++ks) {               // FULL unroll: Bf in VGPRs
      v2f a = *(const v2f*)(ap + (ks << 2));        // b64 load, imm offsets
      c = __builtin_amdgcn_wmma_f32_16x16x4_f32(
              /*neg_a=*/false, a, /*neg_b=*/false, Bf[ks],
              /*c_mod=*/(short)0, c, /*reuse_a=*/false, /*reuse_b=*/false);
    }

    float* yp = Y + (size_t)(rowBase + mh) * FAN + ncol;
    if (rowBase + 16 <= num_rows) {
      // Fast path: one address, 8 stores with immediate offsets (r*512B).
#pragma unroll
      for (int r = 0; r < 8; ++r) yp[(size_t)r * FAN] = c[r];
    } else {
#pragma unroll
      for (int r = 0; r < 8; ++r)
        if (rowBase + mh + r < num_rows) yp[(size_t)r * FAN] = c[r];
    }
  }
}

// ---------------------------------------------------------------------------
// conv[seg_i[t]] += w[t] * f[idx_j[t]]   (seg_i sorted)
// One wave per contiguous chunk of interactions; each lane owns 4 channels
// (b128 loads -> one coalesced 512B line of w per interaction per wave).
// Register accumulation until the segment id changes (avg run length 16),
// then flush with hardware fp32 global atomics -> ~16x fewer atomics.
// ---------------------------------------------------------------------------
__global__ __launch_bounds__(256) void cfconv_scatter_kernel(
    const float* __restrict__ w,
    const float* __restrict__ f,
    const int*   __restrict__ seg_i,
    const int*   __restrict__ idx_j,
    float*       __restrict__ conv,
    int num_inter, int chunk)
{
  const int lane = threadIdx.x & 31;
  int wid = (blockIdx.x << 3) + (threadIdx.x >> 5);
  wid = __builtin_amdgcn_readfirstlane(wid);         // wave-uniform -> s_loads
  const long long start = (long long)wid * chunk;
  if (start >= num_inter) return;
  long long end = start + chunk;
  if (end > num_inter) end = num_inter;
  const int c0 = lane << 2;                          // 4 channels per lane

  v4f acc = {0.0f, 0.0f, 0.0f, 0.0f};
  int curSeg = seg_i[start];

  for (long long it = start; it < end; ++it) {
    const int seg = seg_i[it];
    const int j   = idx_j[it];
    if (seg != curSeg) {
      float* dst = conv + (size_t)curSeg * FAN + c0;
      unsafeAtomicAdd(dst + 0, acc.x);
      unsafeAtomicAdd(dst + 1, acc.y);
      unsafeAtomicAdd(dst + 2, acc.z);
      unsafeAtomicAdd(dst + 3, acc.w);
      acc = (v4f){0.0f, 0.0f, 0.0f, 0.0f};
      curSeg = seg;
    }
    __builtin_prefetch(w + (it + 8) * FAN + c0, 0, 0);  // global_prefetch_b8
    const v4f wv = *(const v4f*)(w + (size_t)it * FAN + c0);
    const v4f fv = *(const v4f*)(f + (size_t)j  * FAN + c0);  // L2-resident
    acc += wv * fv;
  }
  float* dst = conv + (size_t)curSeg * FAN + c0;
  unsafeAtomicAdd(dst + 0, acc.x);
  unsafeAtomicAdd(dst + 1, acc.y);
  unsafeAtomicAdd(dst + 2, acc.z);
  unsafeAtomicAdd(dst + 3, acc.w);
}

// ---------------------------------------------------------------------------
extern "C" void kernel_launch(void* const* d_in, const int* in_sizes, int n_in,
                              void* d_out, int out_size, void* d_ws, size_t ws_size,
                              hipStream_t stream) {
  (void)n_in; (void)out_size; (void)ws_size;
  const float* x         = (const float*)d_in[0];
  const float* w         = (const float*)d_in[1];
  const int*   seg_i     = (const int*)  d_in[2];
  const int*   idx_j     = (const int*)  d_in[3];
  const float* W_in2fac  = (const float*)d_in[4];
  const float* W_fac2out = (const float*)d_in[5];
  const float* b_fac2out = (const float*)d_in[6];
  float*       y         = (float*)d_out;

  const int num_atoms = in_sizes[0] / FAN;
  const int num_inter = in_sizes[1] / FAN;

  float* f    = (float*)d_ws;                         // num_atoms x 128
  float* conv = f + (size_t)num_atoms * FAN;          // num_atoms x 128

  // 1) conv = 0
  zero_kernel<<<dim3(1024), dim3(256), 0, stream>>>(conv, (long long)num_atoms * FAN);

  // 2) f = x @ W_in2fac   (fp32 WMMA; f stays hot in 192MB L2 for the gather)
  const int rowBlocks = (num_atoms + 15) / 16;
  const int g1 = rowBlocks < 1568 ? rowBlocks : 1568;
  gemm128_wmma_kernel<<<dim3(g1), dim3(256), 0, stream>>>(x, W_in2fac, nullptr, f, num_atoms);

  // 3) conv = segment_sum(w * f[idx_j], seg_i)  -- HBM-bound phase (~820MB of w)
  const int chunk  = 256;
  const int waves  = (num_inter + chunk - 1) / chunk;
  const int blocks = (waves + 7) / 8;
  cfconv_scatter_kernel<<<dim3(blocks), dim3(256), 0, stream>>>(
      w, f, seg_i, idx_j, conv, num_inter, chunk);

  // 4) y = conv @ W_fac2out + b
  gemm128_wmma_kernel<<<dim3(g1), dim3(256), 0, stream>>>(conv, W_fac2out, b_fac2out, y, num_atoms);
}